// AttnDecoderRNN2_40767829574385
// MI455X (gfx1250) — compile-verified
//
#include <hip/hip_runtime.h>
#include <hip/hip_bf16.h>
#include <math.h>

typedef __attribute__((ext_vector_type(2))) float v2f;
typedef __attribute__((ext_vector_type(8))) float v8f;

#define ACT_NONE 0
#define ACT_RELU 1
#define ACT_TANH 2
#define ACT_SSIGN 3

// Y[M,N] = act( X[M,K] @ W[N,K]^T + b0 + b1 )
// grid.x = N/16 (one wave per 16-col tile), grid.y = M/64 (wave covers 4 M-subtiles)
// WIN=1: X rows are the 10-step attention window of input_enc:
//        row m -> input_enc row (m/10)*2000 + (m%10)
// A frag (16x4 f32): lane&15 = M row, lane>>4 selects K pair {0,1}/{2,3}.
// B frag (4x16 f32): lane&15 = N col, same K-pair striping.
// C/D (16x16 f32, 8 VGPRs): m = vgpr + 8*(lane>>4), n = lane&15.
// Requires K % 16 == 0 (true for all GEMMs here).
template <int ACT, int WIN>
__global__ __launch_bounds__(32) void gemm_xwt_wmma(
    const float* __restrict__ X, const float* __restrict__ W,
    const float* __restrict__ b0, const float* __restrict__ b1,
    float* __restrict__ Y, int K, int N) {
  const int lane = threadIdx.x & 31;
  const int idx  = lane & 15;   // M row / N col within tile
  const int hk   = lane >> 4;   // K-pair selector
  const int n0   = blockIdx.x * 16;
  const int m0   = blockIdx.y * 64;

  v8f acc[4];
#pragma unroll
  for (int mt = 0; mt < 4; ++mt)
#pragma unroll
    for (int i = 0; i < 8; ++i) acc[mt][i] = 0.0f;

  const float* wrow = W + (size_t)(n0 + idx) * K;
  const float* xrow[4];
#pragma unroll
  for (int mt = 0; mt < 4; ++mt) {
    const int m = m0 + mt * 16 + idx;
    const size_t r = WIN ? ((size_t)(m / 10) * 2000 + (size_t)(m % 10)) : (size_t)m;
    xrow[mt] = X + r * (size_t)K;
  }

  for (int k = 0; k < K; k += 16) {
    // Keep the streamed weight row ahead of the WMMA consumer.
    __builtin_prefetch(wrow + k + 128, 0, 1);
    // ---- load phase: 4 B fragments + 16 A fragments (b64 each) ----
    v2f bf[4];
    v2f af[4][4];
#pragma unroll
    for (int u = 0; u < 4; ++u) {
      const int kb = k + 4 * u + hk * 2;
      bf[u] = *(const v2f*)(wrow + kb);
#pragma unroll
      for (int mt = 0; mt < 4; ++mt) af[mt][u] = *(const v2f*)(xrow[mt] + kb);
    }
    // ---- math phase: 16 WMMAs, waits can now be partial/overlapped ----
#pragma unroll
    for (int u = 0; u < 4; ++u)
#pragma unroll
      for (int mt = 0; mt < 4; ++mt)
        acc[mt] = __builtin_amdgcn_wmma_f32_16x16x4_f32(
            false, af[mt][u], false, bf[u], (short)0, acc[mt], false, false);
  }

  const int n = n0 + idx;
  float bias = 0.0f;
  if (b0) bias += b0[n];
  if (b1) bias += b1[n];
#pragma unroll
  for (int mt = 0; mt < 4; ++mt) {
#pragma unroll
    for (int v = 0; v < 8; ++v) {
      const int m = m0 + mt * 16 + hk * 8 + v;
      float val = acc[mt][v] + bias;
      if (ACT == ACT_RELU) val = fmaxf(val, 0.0f);
      else if (ACT == ACT_TANH) val = tanhf(val);
      else if (ACT == ACT_SSIGN) val = val / (1.0f + fabsf(val));
      Y[(size_t)m * N + n] = val;
    }
  }
}

// x1[64,144] = concat(input_dec[64,80], spkr[64,64])
__global__ void pack_x1(const float* __restrict__ dec, const float* __restrict__ spkr,
                        float* __restrict__ x1) {
  const int n = blockIdx.x, j = threadIdx.x;
  if (j < 144) x1[n * 144 + j] = (j < 80) ? dec[n * 80 + j] : spkr[n * 64 + (j - 80)];
}

// sp1[64,256] = relu(speed[n]*W_sp1 + b_sp1)
__global__ void sp1_kernel(const float* __restrict__ speed, const float* __restrict__ W_sp1,
                           const float* __restrict__ b_sp1, float* __restrict__ out) {
  const int n = blockIdx.x, j = threadIdx.x;
  out[n * 256 + j] = fmaxf(speed[n] * W_sp1[j] + b_sp1[j], 0.0f);
}

// Per-batch energies over the 10-step window -> normalized attention weights.
__global__ void energy_att(const float* __restrict__ aenc, const float* __restrict__ aspk,
                           const float* __restrict__ conv_prev,
                           const float* __restrict__ W_speed_att,
                           const float* __restrict__ W_proj, const float* __restrict__ b_proj,
                           const float* __restrict__ speed, const int* __restrict__ lengths,
                           float* __restrict__ attw) {
  __shared__ float red[256];
  __shared__ float logits[10];
  const int n = blockIdx.x, a = threadIdx.x;
  const float asp = aspk[n * 256 + a];
  const float spd = speed[n] * W_speed_att[a];
  const float wp  = W_proj[a];
  for (int t = 0; t < 10; ++t) {
    // in_att_prev = conv(one-hot@t=0) = conv_prev[a, 15 - t]
    float e = aenc[(size_t)(n * 10 + t) * 256 + a] + asp + conv_prev[a * 31 + (15 - t)] + spd;
    red[a] = tanhf(e) * wp;
    __syncthreads();
    for (int s = 128; s > 0; s >>= 1) {
      if (a < s) red[a] += red[a + s];
      __syncthreads();
    }
    if (a == 0) logits[t] = red[0] + b_proj[0];
    __syncthreads();
  }
  if (a == 0) {
    const int len  = lengths[n];
    const int len1 = (len > 1 ? len : 1) - 1;
    const int hi   = len1 < 9 ? len1 : 9;     // window [0, hi]
    float m = -3.4e38f;
    for (int t = 0; t <= hi; ++t) m = fmaxf(m, logits[t]);
    float w[10];
    float s = 0.0f;
    for (int t = 0; t < 10; ++t) {
      w[t] = (t <= hi) ? expf(logits[t] - m) : 0.0f;
      s += w[t];
    }
    const float inv = 1.0f / fmaxf(s, 1e-12f);
    float tot = 0.0f;
    for (int t = 0; t < 10; ++t) {
      const float aw = w[t] * inv;
      attw[n * 16 + t] = aw;
      tot += aw;
    }
    attw[n * 16 + 10] = tot;
  }
}

// context[n,e] = sum_t att[n,t]*input_enc[n,t,e] + sp[n,e]*sum_t att[n,t]
__global__ void context_kernel(const float* __restrict__ enc, const float* __restrict__ sp,
                               const float* __restrict__ attw, float* __restrict__ ctx,
                               float* __restrict__ out_ctx) {
  const int n = blockIdx.x;
  float aw[10];
  for (int t = 0; t < 10; ++t) aw[t] = attw[n * 16 + t];
  const float tot = attw[n * 16 + 10];
  for (int e = threadIdx.x; e < 512; e += blockDim.x) {
    float acc = sp[n * 512 + e] * tot;
    for (int t = 0; t < 10; ++t) acc += aw[t] * enc[((size_t)n * 2000 + t) * 512 + e];
    ctx[n * 512 + e] = acc;
    out_ctx[n * 512 + e] = acc;
  }
}

// x2[64,832] = [pre(256) | ctx(512) | spkr(64)]
__global__ void pack_x2(const float* __restrict__ pre, const float* __restrict__ ctx,
                        const float* __restrict__ spkr, float* __restrict__ x2) {
  const int n = blockIdx.x;
  for (int j = threadIdx.x; j < 832; j += blockDim.x) {
    float v;
    if (j < 256) v = pre[n * 256 + j];
    else if (j < 768) v = ctx[n * 512 + (j - 256)];
    else v = spkr[n * 64 + (j - 768)];
    x2[n * 832 + j] = v;
  }
}

// LSTM gate nonlinearity with zero initial state: h = sig(o)*tanh(sig(i)*tanh(g))
__global__ void lstm_act(const float* __restrict__ g, float* __restrict__ h) {
  const int n = blockIdx.x;
  const float* gr = g + (size_t)n * 4096;
  for (int j = threadIdx.x; j < 1024; j += blockDim.x) {
    const float i  = gr[j];
    const float gg = gr[2048 + j];
    const float o  = gr[3072 + j];
    const float si = 1.0f / (1.0f + expf(-i));
    const float so = 1.0f / (1.0f + expf(-o));
    const float c  = si * tanhf(gg);
    h[n * 1024 + j] = so * tanhf(c);
  }
}

// x3[64,1536] = [h2(1024) | ctx(512)]
__global__ void pack_x3(const float* __restrict__ h2, const float* __restrict__ ctx,
                        float* __restrict__ x3) {
  const int n = blockIdx.x;
  for (int j = threadIdx.x; j < 1536; j += blockDim.x)
    x3[n * 1536 + j] = (j < 1024) ? h2[n * 1024 + j] : ctx[n * 512 + (j - 1024)];
}

extern "C" void kernel_launch(void* const* d_in, const int* in_sizes, int n_in,
                              void* d_out, int out_size, void* d_ws, size_t ws_size,
                              hipStream_t stream) {
  const float* input_enc   = (const float*)d_in[0];
  const float* input_dec   = (const float*)d_in[1];
  const float* spkr_vec    = (const float*)d_in[2];   // [64,64]
  const int*   lengths_enc = (const int*)d_in[3];
  const float* speed       = (const float*)d_in[4];
  const float* W_enc       = (const float*)d_in[5];
  const float* b_enc       = (const float*)d_in[6];
  const float* W_spkr      = (const float*)d_in[7];
  const float* conv_prev   = (const float*)d_in[8];
  const float* W_speed_att = (const float*)d_in[9];
  const float* W_proj      = (const float*)d_in[10];
  const float* b_proj      = (const float*)d_in[11];
  const float* W_sp1       = (const float*)d_in[12];
  const float* b_sp1       = (const float*)d_in[13];
  const float* W_sp2       = (const float*)d_in[14];
  const float* b_sp2       = (const float*)d_in[15];
  const float* W_p1        = (const float*)d_in[16];
  const float* b_p1        = (const float*)d_in[17];
  const float* W_p2        = (const float*)d_in[18];
  const float* b_p2        = (const float*)d_in[19];
  const float* Wih0        = (const float*)d_in[20];
  // d_in[21] Whh0: unused (initial h = 0)
  const float* bih0        = (const float*)d_in[22];
  const float* bhh0        = (const float*)d_in[23];
  const float* Wih1        = (const float*)d_in[24];
  // d_in[25] Whh1: unused (initial h = 0)
  const float* bih1        = (const float*)d_in[26];
  const float* bhh1        = (const float*)d_in[27];
  const float* W_out       = (const float*)d_in[28];
  const float* b_out       = (const float*)d_in[29];

  float* out      = (float*)d_out;       // [64,2,80] flat = [64,160]
  float* out_ctx  = out + 64 * 160;      // context [64,512]

  float* ws = (float*)d_ws;
  size_t off = 0;
  float* aenc = ws + off; off += 640 * 256;  // softsign(enc_window @ W_enc^T + b)
  float* x1   = ws + off; off += 64 * 144;
  float* t1   = ws + off; off += 64 * 512;
  float* pre  = ws + off; off += 64 * 256;
  float* sp1b = ws + off; off += 64 * 256;
  float* spb  = ws + off; off += 64 * 512;
  float* aspk = ws + off; off += 64 * 256;
  float* attw = ws + off; off += 64 * 16;
  float* ctx  = ws + off; off += 64 * 512;
  float* x2   = ws + off; off += 64 * 832;
  float* gbuf = ws + off; off += 64 * 4096;
  float* h1   = ws + off; off += 64 * 1024;
  float* h2   = ws + off; off += 64 * 1024;
  float* x3   = ws + off; off += 64 * 1536;
  (void)ws_size; (void)n_in; (void)in_sizes; (void)out_size;

  // --- attention-bias path (only window t=0..9 matters; WIN GEMM reads in place) ---
  gemm_xwt_wmma<ACT_SSIGN, 1><<<dim3(16, 10), 32, 0, stream>>>(input_enc, W_enc, b_enc, nullptr, aenc, 512, 256);
  gemm_xwt_wmma<ACT_SSIGN, 0><<<dim3(16, 1), 32, 0, stream>>>(spkr_vec, W_spkr, nullptr, nullptr, aspk, 64, 256);

  // --- prenet ---
  pack_x1<<<64, 160, 0, stream>>>(input_dec, spkr_vec, x1);
  gemm_xwt_wmma<ACT_RELU, 0><<<dim3(32, 1), 32, 0, stream>>>(x1, W_p1, b_p1, nullptr, t1, 144, 512);
  gemm_xwt_wmma<ACT_RELU, 0><<<dim3(16, 1), 32, 0, stream>>>(t1, W_p2, b_p2, nullptr, pre, 512, 256);

  // --- speed projection ---
  sp1_kernel<<<64, 256, 0, stream>>>(speed, W_sp1, b_sp1, sp1b);
  gemm_xwt_wmma<ACT_TANH, 0><<<dim3(32, 1), 32, 0, stream>>>(sp1b, W_sp2, b_sp2, nullptr, spb, 256, 512);

  // --- energies, attention weights, context ---
  energy_att<<<64, 256, 0, stream>>>(aenc, aspk, conv_prev, W_speed_att, W_proj, b_proj,
                                     speed, lengths_enc, attw);
  context_kernel<<<64, 256, 0, stream>>>(input_enc, spb, attw, ctx, out_ctx);

  // --- 2-layer LSTM step (zero initial state => recurrent GEMMs vanish) ---
  pack_x2<<<64, 256, 0, stream>>>(pre, ctx, spkr_vec, x2);
  gemm_xwt_wmma<ACT_NONE, 0><<<dim3(256, 1), 32, 0, stream>>>(x2, Wih0, bih0, bhh0, gbuf, 832, 4096);
  lstm_act<<<64, 256, 0, stream>>>(gbuf, h1);
  gemm_xwt_wmma<ACT_NONE, 0><<<dim3(256, 1), 32, 0, stream>>>(h1, Wih1, bih1, bhh1, gbuf, 1024, 4096);
  lstm_act<<<64, 256, 0, stream>>>(gbuf, h2);

  // --- output projection ---
  pack_x3<<<64, 256, 0, stream>>>(h2, ctx, x3);
  gemm_xwt_wmma<ACT_NONE, 0><<<dim3(10, 1), 32, 0, stream>>>(x3, W_out, b_out, nullptr, out, 1536, 160);
}